// GraphQNetwork_19653770347252
// MI455X (gfx1250) — compile-verified
//
#include <hip/hip_runtime.h>

// GraphQNetwork on MI455X (gfx1250).
// Bandwidth-bound workload -> keep everything f32 (exact vs reference),
// use V_WMMA_F32_16X16X4_F32 for the two dense GEMMs (conv2 linear, fc1).
// B-matrix is staged into LDS pre-swizzled so each lane's (b.x,b.y) WMMA
// fragment is one contiguous ds_load_b64 (no VALU re-packing in the loop).

#define N_NODES   262144
#define NPG       256
#define N_GRAPHS  1024
#define N_EDGES   2097152
#define NF        9
#define NS        8
#define C1        32
#define C2        64
#define NROWS     (N_GRAPHS * NS)   // 8192 station rows
#define CCOMB     128

typedef __attribute__((ext_vector_type(2))) float v2f;
typedef __attribute__((ext_vector_type(8))) float v8f;

// ---------------- utility ----------------
__global__ void k_zero(float* __restrict__ p, long long n) {
  long long i = (long long)blockIdx.x * blockDim.x + threadIdx.x;
  long long stride = (long long)gridDim.x * blockDim.x;
  for (; i < n; i += stride) p[i] = 0.0f;
}

// ---------------- degree / normalization ----------------
__global__ void k_deg(const int* __restrict__ dst, float* __restrict__ deg) {
  int e = blockIdx.x * blockDim.x + threadIdx.x;
  if (e < N_EDGES) atomicAdd(&deg[dst[e]], 1.0f);
}

__global__ void k_dinv(float* __restrict__ deg) {
  int i = blockIdx.x * blockDim.x + threadIdx.x;
  if (i < N_NODES) deg[i] = rsqrtf(deg[i] + 1.0f);  // +1 self loop
}

// ---------------- conv1 linear: h1lin = x @ W1 (9x32, VALU) ----------------
__global__ void k_lin1(const float* __restrict__ x, const float* __restrict__ W1,
                       float* __restrict__ h1lin) {
  __shared__ float w[NF * C1];
  for (int i = threadIdx.x; i < NF * C1; i += blockDim.x) w[i] = W1[i];
  __syncthreads();
  int node = blockIdx.x * blockDim.x + threadIdx.x;
  float xi[NF];
#pragma unroll
  for (int f = 0; f < NF; ++f) xi[f] = x[(size_t)node * NF + f];
#pragma unroll
  for (int c = 0; c < C1; ++c) {
    float s = 0.0f;
#pragma unroll
    for (int f = 0; f < NF; ++f) s = fmaf(xi[f], w[f * C1 + c], s);
    h1lin[(size_t)node * C1 + c] = s;
  }
}

// ---------------- edge scatter: agg[dst] += norm * hlin[src] ----------------
template <int NCH>
__global__ void k_scat(const int* __restrict__ src, const int* __restrict__ dst,
                       const float* __restrict__ dinv, const float* __restrict__ hlin,
                       float* __restrict__ agg) {
  long long stride = (long long)gridDim.x * blockDim.x;
  for (long long e = (long long)blockIdx.x * blockDim.x + threadIdx.x; e < N_EDGES;
       e += stride) {
    int s = src[e], d = dst[e];
    if (e + stride < N_EDGES)  // hide gather latency of next iteration
      __builtin_prefetch(&hlin[(size_t)src[e + stride] * NCH], 0, 0);
    float norm = dinv[s] * dinv[d];
    const float4* hp = (const float4*)(hlin + (size_t)s * NCH);
    float* ap = agg + (size_t)d * NCH;
#pragma unroll
    for (int q = 0; q < NCH / 4; ++q) {
      float4 v = hp[q];
      atomicAdd(ap + 4 * q + 0, norm * v.x);
      atomicAdd(ap + 4 * q + 1, norm * v.y);
      atomicAdd(ap + 4 * q + 2, norm * v.z);
      atomicAdd(ap + 4 * q + 3, norm * v.w);
    }
  }
}

// ---------------- self-loop + bias + ReLU (in place into agg) ----------------
template <int NCH>
__global__ void k_post(const float* __restrict__ dinv, const float* __restrict__ hlin,
                       const float* __restrict__ bias, float* __restrict__ agg) {
  int i = blockIdx.x * blockDim.x + threadIdx.x;  // < N_NODES*NCH
  int node = i / NCH, c = i % NCH;
  float di = dinv[node];
  float v = agg[i] + di * di * hlin[i] + bias[c];
  agg[i] = v > 0.0f ? v : 0.0f;
}

// Stage a [K x 64] row-major weight matrix into LDS, swizzled so that the
// WMMA B fragment for (k-step, lane-half, n-tile, lane) is one contiguous
// float2:  slot(k,half,n,l16) = ((k*2+half)*4 + n)*16 + l16, 2 floats/slot,
// holding W[(4k+2*half)+0][n*16+l16] and W[(4k+2*half)+1][n*16+l16].
template <int K>
__device__ inline void stage_B_swizzled(const float* __restrict__ W,
                                        float* __restrict__ lds) {
  const int slots = (K / 2) * 64;  // (K/4)*2*4*16
  for (int s = threadIdx.x; s < slots; s += blockDim.x) {
    int l16 = s & 15;
    int n = (s >> 4) & 3;
    int kh = s >> 6;  // k*2 + half  in [0, K/2)
    int r = kh * 2;   // 4k + 2*half
    lds[s * 2 + 0] = W[(r + 0) * C2 + n * 16 + l16];
    lds[s * 2 + 1] = W[(r + 1) * C2 + n * 16 + l16];
  }
}

// ---------------- WMMA f32 GEMM: h2lin[N,64] = h1[N,32] @ W2[32,64] ----------
// One wave computes one 16x64 output strip via 8 K-steps x 4 N-tiles of
// V_WMMA_F32_16X16X4_F32. A fragment: lane(L) row = L&15, K pair = 4k+2*(L>>4).
__global__ void k_gemm_conv2(const float* __restrict__ h1, const float* __restrict__ W2,
                             float* __restrict__ h2lin) {
  __shared__ float w[C1 * C2];  // 8 KB, swizzled
  stage_B_swizzled<C1>(W2, w);
  __syncthreads();
  int wave = threadIdx.x >> 5, lane = threadIdx.x & 31;
  int tile = blockIdx.x * 8 + wave;  // 16-row tile index
  int l16 = lane & 15, half = lane >> 4;
  int row = tile * 16 + l16;
  const float* bbase = w + (size_t)(half * 64 + l16) * 2;  // per-lane base
  v8f acc[4] = {};
#pragma unroll
  for (int k = 0; k < C1 / 4; ++k) {
    v2f a = *(const v2f*)(h1 + (size_t)row * C1 + 4 * k + 2 * half);
#pragma unroll
    for (int n = 0; n < 4; ++n) {
      v2f b = *(const v2f*)(bbase + (k * 128 + n * 16) * 2);
      acc[n] = __builtin_amdgcn_wmma_f32_16x16x4_f32(
          false, a, false, b, (short)0, acc[n], false, false);
    }
  }
#pragma unroll
  for (int n = 0; n < 4; ++n)
#pragma unroll
    for (int v = 0; v < 8; ++v)
      h2lin[(size_t)(tile * 16 + v + 8 * half) * C2 + n * 16 + l16] = acc[n][v];
}

// ---------------- per-graph mean pool over 256 nodes -------------------------
__global__ void k_pool(const float* __restrict__ h2, float* __restrict__ ctx) {
  int g = blockIdx.x;
  int c = threadIdx.x & 63;
  int part = threadIdx.x >> 6;  // 0..3
  float s = 0.0f;
  for (int n = part; n < NPG; n += 4) s += h2[((size_t)g * NPG + n) * C2 + c];
  __shared__ float red[256];
  red[threadIdx.x] = s;
  __syncthreads();
  if (part == 0) {
    float t = red[c] + red[64 + c] + red[128 + c] + red[192 + c];
    ctx[g * C2 + c] = t * (1.0f / NPG);
  }
}

// ---------------- combined[B*S,128] = [h2[station] , ctx[graph]] -------------
__global__ void k_comb(const float* __restrict__ h2, const float* __restrict__ ctx,
                       const int* __restrict__ stations, float* __restrict__ comb) {
  int i = blockIdx.x * blockDim.x + threadIdx.x;  // < NROWS*CCOMB
  int r = i >> 7, c = i & 127;
  int b = r >> 3, s = r & 7;
  float v;
  if (c < C2)
    v = h2[((size_t)b * NPG + stations[s]) * C2 + c];
  else
    v = ctx[b * C2 + (c - C2)];
  comb[i] = v;
}

// ---------------- WMMA f32 GEMM: hfc = relu(comb[8192,128]@fc1W[128,64]+b) ---
__global__ void k_gemm_fc1(const float* __restrict__ comb, const float* __restrict__ fc1W,
                           const float* __restrict__ fc1b, float* __restrict__ hfc) {
  __shared__ float w[CCOMB * C2];  // 32 KB, swizzled
  stage_B_swizzled<CCOMB>(fc1W, w);
  __syncthreads();
  int wave = threadIdx.x >> 5, lane = threadIdx.x & 31;
  int tile = blockIdx.x * 8 + wave;  // 512 tiles total
  int l16 = lane & 15, half = lane >> 4;
  int row = tile * 16 + l16;
  const float* bbase = w + (size_t)(half * 64 + l16) * 2;
  v8f acc[4] = {};
#pragma unroll
  for (int k = 0; k < CCOMB / 4; ++k) {
    v2f a = *(const v2f*)(comb + (size_t)row * CCOMB + 4 * k + 2 * half);
#pragma unroll
    for (int n = 0; n < 4; ++n) {
      v2f b = *(const v2f*)(bbase + (k * 128 + n * 16) * 2);
      acc[n] = __builtin_amdgcn_wmma_f32_16x16x4_f32(
          false, a, false, b, (short)0, acc[n], false, false);
    }
  }
#pragma unroll
  for (int n = 0; n < 4; ++n) {
    float bias = fc1b[n * 16 + l16];
#pragma unroll
    for (int v = 0; v < 8; ++v) {
      float z = acc[n][v] + bias;
      hfc[(size_t)(tile * 16 + v + 8 * half) * C2 + n * 16 + l16] = z > 0.0f ? z : 0.0f;
    }
  }
}

// ---------------- fc2: q[r] = dot(hfc[r], fc2W) + fc2b -----------------------
__global__ void k_fc2(const float* __restrict__ hfc, const float* __restrict__ fc2W,
                      const float* __restrict__ fc2b, float* __restrict__ out) {
  int r = blockIdx.x * blockDim.x + threadIdx.x;
  if (r >= NROWS) return;
  float s = fc2b[0];
  const float4* hp = (const float4*)(hfc + (size_t)r * C2);
  const float4* wp = (const float4*)fc2W;
#pragma unroll
  for (int q = 0; q < C2 / 4; ++q) {
    float4 h = hp[q], w = wp[q];
    s = fmaf(h.x, w.x, s);
    s = fmaf(h.y, w.y, s);
    s = fmaf(h.z, w.z, s);
    s = fmaf(h.w, w.w, s);
  }
  out[r] = s;
}

// ---------------- launcher ----------------
extern "C" void kernel_launch(void* const* d_in, const int* in_sizes, int n_in,
                              void* d_out, int out_size, void* d_ws, size_t ws_size,
                              hipStream_t stream) {
  const float* x        = (const float*)d_in[0];
  const int*   eidx     = (const int*)d_in[1];  // [2, E] stacked
  const int*   stations = (const int*)d_in[2];
  const float* W1   = (const float*)d_in[3];
  const float* b1   = (const float*)d_in[4];
  const float* W2   = (const float*)d_in[5];
  const float* b2   = (const float*)d_in[6];
  const float* fc1W = (const float*)d_in[7];
  const float* fc1b = (const float*)d_in[8];
  const float* fc2W = (const float*)d_in[9];
  const float* fc2b = (const float*)d_in[10];
  float* out = (float*)d_out;

  const int* src = eidx;
  const int* dst = eidx + N_EDGES;

  // Workspace layout (floats); regions reused once their producer phase ends.
  float* ws    = (float*)d_ws;
  float* dinv  = ws;                                   // [N]
  float* bufA  = ws + N_NODES;                         // [64N]
  float* h1lin = bufA;                                 //   [0,32N)
  float* agg1  = bufA + (size_t)N_NODES * C1;          //   [32N,64N) -> h1
  float* agg2  = bufA;                                 //   reused [0,64N) -> h2
  float* bufB  = bufA + (size_t)N_NODES * C2;          // [64N]
  float* h2lin = bufB;                                 //   conv2 linear out
  float* comb  = bufB;                                 //   reused: [8192,128]
  float* hfc   = bufB + (size_t)NROWS * CCOMB;         //   [8192,64]
  float* ctx   = bufB + (size_t)N_NODES * C2;          // [1024*64]

  // --- conv1 ---
  k_zero<<<1024, 256, 0, stream>>>(dinv, (long long)N_NODES);
  k_zero<<<4096, 256, 0, stream>>>(agg1, (long long)N_NODES * C1);
  k_deg<<<N_EDGES / 256, 256, 0, stream>>>(dst, dinv);
  k_dinv<<<N_NODES / 256, 256, 0, stream>>>(dinv);
  k_lin1<<<N_NODES / 256, 256, 0, stream>>>(x, W1, h1lin);
  k_scat<C1><<<4096, 256, 0, stream>>>(src, dst, dinv, h1lin, agg1);
  k_post<C1><<<(N_NODES * C1) / 256, 256, 0, stream>>>(dinv, h1lin, b1, agg1);

  // --- conv2 (WMMA GEMM then scatter) ---
  k_gemm_conv2<<<N_NODES / 128, 256, 0, stream>>>(agg1, W2, h2lin);
  k_zero<<<4096, 256, 0, stream>>>(agg2, (long long)N_NODES * C2);
  k_scat<C2><<<4096, 256, 0, stream>>>(src, dst, dinv, h2lin, agg2);
  k_post<C2><<<(N_NODES * C2) / 256, 256, 0, stream>>>(dinv, h2lin, b2, agg2);

  // --- pooling + station head ---
  k_pool<<<N_GRAPHS, 256, 0, stream>>>(agg2, ctx);
  k_comb<<<(NROWS * CCOMB) / 256, 256, 0, stream>>>(agg2, ctx, stations, comb);
  k_gemm_fc1<<<NROWS / 128, 256, 0, stream>>>(comb, fc1W, fc1b, hfc);
  k_fc2<<<NROWS / 256, 256, 0, stream>>>(hfc, fc2W, fc2b, out);
}